// Radon_47218870452277
// MI455X (gfx1250) — compile-verified
//
#include <hip/hip_runtime.h>

#define SIZE 512
#define NA 180
#define BATCH 32
#define NB 16     // images accumulated per thread (batch-amortized weights)
#define NGRP (BATCH / NB)
#define WPB 256   // threads per block = w-columns per block
#define PACKED_BYTES ((size_t)SIZE * SIZE * BATCH * sizeof(float))

typedef float v2f __attribute__((ext_vector_type(2)));

__device__ __forceinline__ void pkfma(v2f& d, v2f a, v2f b) {
  // d = a * b + d  (two f32 FMAs in one VOP3P instruction)
  asm("v_pk_fma_f32 %0, %1, %2, %0" : "+v"(d) : "v"(a), "v"(b));
}

// ---------------------------------------------------------------------------
// Pass 1: repack NCHW (batch-outer, 1 MB stride) into batch-innermost tiles:
//   packed[((g*SIZE + y)*SIZE + x)*NB + j] = img[(g*NB + j)][y][x]
// Streaming, fully coalesced reads per j; 64 B contiguous write per pixel.
// ---------------------------------------------------------------------------
__global__ __launch_bounds__(256) void Radon_pack_kernel(
    const float* __restrict__ img, float* __restrict__ packed) {
  const int idx = blockIdx.x * 256 + threadIdx.x;   // over NGRP*SIZE*SIZE
  const int x = idx & (SIZE - 1);
  const int y = (idx >> 9) & (SIZE - 1);
  const int g = idx >> 18;

  float vals[NB];
#pragma unroll
  for (int j = 0; j < NB; ++j)
    vals[j] = img[((size_t)(g * NB + j) * SIZE + y) * SIZE + x];

  float4* dst = (float4*)(packed + (size_t)idx * NB);
#pragma unroll
  for (int q = 0; q < NB / 4; ++q)
    dst[q] = make_float4(vals[4 * q], vals[4 * q + 1], vals[4 * q + 2],
                         vals[4 * q + 3]);
}

// ---------------------------------------------------------------------------
// Per-h bilinear setup: 4 combined tap weights + 2 row offsets, branchless
// edge handling identical to the reference's validity masking.
// ---------------------------------------------------------------------------
struct TapSetup {
  v2f W00, W01, W10, W11;
  size_t o00, o10;
};

__device__ __forceinline__ TapSetup tap_setup(float ix, float iy) {
  const float x0f = floorf(ix);
  const float y0f = floorf(iy);
  const float fx  = ix - x0f;
  const float fy  = iy - y0f;
  const int   x0  = (int)x0f;
  const int   y0  = (int)y0f;

  const bool vx0 = ((unsigned)x0       < (unsigned)SIZE);
  const bool vx1 = ((unsigned)(x0 + 1) < (unsigned)SIZE);
  const bool vy0 = ((unsigned)y0       < (unsigned)SIZE);
  const bool vy1 = ((unsigned)(y0 + 1) < (unsigned)SIZE);

  float wx0 = vx0 ? (1.0f - fx) : 0.0f;
  float wx1 = vx1 ? fx          : 0.0f;
  const float wy0 = vy0 ? (1.0f - fy) : 0.0f;
  const float wy1 = vy1 ? fy          : 0.0f;

  // Load columns {xb, xb+1}, clamped in-bounds; remap x-tap weights onto the
  // loaded lanes.
  const int xb = min(max(x0, 0), SIZE - 2);
  float g0 = wx0, g1 = wx1;
  if (x0 < 0)             { g0 = wx1; g1 = 0.0f; }
  else if (x0 > SIZE - 2) { g1 = wx0; g0 = 0.0f; }

  const float c00 = wy0 * g0, c01 = wy0 * g1;
  const float c10 = wy1 * g0, c11 = wy1 * g1;

  const int yb0 = min(max(y0,     0), SIZE - 1);
  const int yb1 = min(max(y0 + 1, 0), SIZE - 1);

  TapSetup t;
  t.W00 = (v2f){c00, c00};
  t.W01 = (v2f){c01, c01};
  t.W10 = (v2f){c10, c10};
  t.W11 = (v2f){c11, c11};
  t.o00 = ((size_t)(yb0 * SIZE + xb)) * NB;
  t.o10 = ((size_t)(yb1 * SIZE + xb)) * NB;
  return t;
}

// ---------------------------------------------------------------------------
// Pass 2: Radon projection from the packed layout.
// Each thread: one (angle, w), 16 images; 512 h-steps along the rotated line.
// Per h: 4 bilinear taps; per tap one 64 B contiguous read (4x b128) serving
// all 16 images; accumulation via v_pk_fma_f32 pairs. h unrolled x2 so two
// independent load clauses are in flight per wave.
// ---------------------------------------------------------------------------
__global__ __launch_bounds__(WPB) void Radon_47218870452277_kernel(
    const float* __restrict__ packed, float* __restrict__ out) {
  const int a = blockIdx.x;                        // angle 0..179
  const int w = blockIdx.y * WPB + threadIdx.x;    // output column 0..511
  const int g = blockIdx.z;                        // batch group (NB images)

  const float th = (float)a * 0.017453292519943295f;
  float s, c;
  __sincosf(th, &s, &c);

  // ix = ( c*lin[w] + s*lin[h] + 1) * 255.5 ; per-h step is exactly (s, c) px
  const float half = 0.5f * (float)(SIZE - 1);
  const float xw   = -1.0f + (2.0f / (float)(SIZE - 1)) * (float)w;
  const float ix0  = ( c * xw - s + 1.0f) * half;
  const float iy0  = (-s * xw - c + 1.0f) * half;

  const float* base = (const float*)__builtin_assume_aligned(
      packed + (size_t)g * SIZE * SIZE * NB, 64);

  v2f acc[NB / 2];
#pragma unroll
  for (int k = 0; k < NB / 2; ++k) acc[k] = (v2f){0.0f, 0.0f};

#pragma unroll 2
  for (int h = 0; h < SIZE; ++h) {
    const float fh = (float)h;
    const TapSetup t = tap_setup(fmaf(fh, s, ix0), fmaf(fh, c, iy0));

#pragma unroll
    for (int q = 0; q < NB / 4; ++q) {
      const float4 t00 = *(const float4*)(base + t.o00      + 4 * q);
      const float4 t01 = *(const float4*)(base + t.o00 + NB + 4 * q);
      const float4 t10 = *(const float4*)(base + t.o10      + 4 * q);
      const float4 t11 = *(const float4*)(base + t.o10 + NB + 4 * q);
      pkfma(acc[2 * q],     t.W00, (v2f){t00.x, t00.y});
      pkfma(acc[2 * q + 1], t.W00, (v2f){t00.z, t00.w});
      pkfma(acc[2 * q],     t.W01, (v2f){t01.x, t01.y});
      pkfma(acc[2 * q + 1], t.W01, (v2f){t01.z, t01.w});
      pkfma(acc[2 * q],     t.W10, (v2f){t10.x, t10.y});
      pkfma(acc[2 * q + 1], t.W10, (v2f){t10.z, t10.w});
      pkfma(acc[2 * q],     t.W11, (v2f){t11.x, t11.y});
      pkfma(acc[2 * q + 1], t.W11, (v2f){t11.z, t11.w});
    }
  }

  // out shape (32, 1, 512, 180): index = n*512*180 + w*180 + a
#pragma unroll
  for (int k = 0; k < NB / 2; ++k) {
    const int n = g * NB + 2 * k;
    out[((size_t)n       * SIZE + (size_t)w) * NA + (size_t)a] = acc[k].x;
    out[((size_t)(n + 1) * SIZE + (size_t)w) * NA + (size_t)a] = acc[k].y;
  }
}

// ---------------------------------------------------------------------------
// Fallback (no workspace): gathers directly from NCHW input.
// ---------------------------------------------------------------------------
#define FNB 8
__global__ __launch_bounds__(WPB) void Radon_fallback_kernel(
    const float* __restrict__ img, float* __restrict__ out) {
  const int a  = blockIdx.x;
  const int w  = blockIdx.y * WPB + threadIdx.x;
  const int n0 = blockIdx.z * FNB;

  const float th = (float)a * 0.017453292519943295f;
  float s, c;
  __sincosf(th, &s, &c);

  const float half = 0.5f * (float)(SIZE - 1);
  const float xw   = -1.0f + (2.0f / (float)(SIZE - 1)) * (float)w;
  const float ix0  = ( c * xw - s + 1.0f) * half;
  const float iy0  = (-s * xw - c + 1.0f) * half;

  const float* base = img + (size_t)n0 * (SIZE * SIZE);

  float acc[FNB];
#pragma unroll
  for (int j = 0; j < FNB; ++j) acc[j] = 0.0f;

#pragma unroll 2
  for (int h = 0; h < SIZE; ++h) {
    const float fh = (float)h;
    const TapSetup t = tap_setup(fmaf(fh, s, ix0), fmaf(fh, c, iy0));
    const float c00 = t.W00.x, c01 = t.W01.x, c10 = t.W10.x, c11 = t.W11.x;
    const size_t o0 = t.o00 / NB;   // scalar-layout offsets
    const size_t o1 = t.o10 / NB;

#pragma unroll
    for (int j = 0; j < FNB; ++j) {
      const float* p = base + (size_t)j * (SIZE * SIZE);
      float v = acc[j];
      v = fmaf(c00, p[o0],     v);
      v = fmaf(c01, p[o0 + 1], v);
      v = fmaf(c10, p[o1],     v);
      v = fmaf(c11, p[o1 + 1], v);
      acc[j] = v;
    }
  }

#pragma unroll
  for (int j = 0; j < FNB; ++j)
    out[((size_t)(n0 + j) * SIZE + (size_t)w) * NA + (size_t)a] = acc[j];
}

extern "C" void kernel_launch(void* const* d_in, const int* in_sizes, int n_in,
                              void* d_out, int out_size, void* d_ws, size_t ws_size,
                              hipStream_t stream) {
  const float* x = (const float*)d_in[0];
  float* out = (float*)d_out;

  if (d_ws != nullptr && ws_size >= PACKED_BYTES) {
    float* packed = (float*)d_ws;
    const int npix = NGRP * SIZE * SIZE;
    Radon_pack_kernel<<<dim3(npix / 256), dim3(256), 0, stream>>>(x, packed);
    dim3 grid(NA, SIZE / WPB, NGRP);  // 180 x 2 x 2
    Radon_47218870452277_kernel<<<grid, dim3(WPB), 0, stream>>>(packed, out);
  } else {
    dim3 grid(NA, SIZE / WPB, BATCH / FNB);  // 180 x 2 x 4
    Radon_fallback_kernel<<<grid, dim3(WPB), 0, stream>>>(x, out);
  }
}